// MemoryMatrixManager_23776938951398
// MI455X (gfx1250) — compile-verified
//
#include <hip/hip_runtime.h>
#include <hip/hip_bf16.h>

// ---------------------------------------------------------------------------
// Chunked linear-attention memory for MI455X (gfx1250, wave32, WMMA + TDM).
// S_t = (S_{t-1} + vhat_t alpha_t^T)/div_t ; o_t = S_t q_t
// Per chunk of L=64 steps, as GEMMs on V_WMMA_F32_16X16X4_F32:
//   W      = Vhat * S            (L x d)   -> m_t = W_t . alpha_t
//   Obase  = Q * S^T             (L x d)   (S staged via double-buffered TDM)
//   Gvv/Gaa/Gqa Gram matrices    (L x L)
//   scalar scan -> div_t, D products, P, g, vscale, 1/D_L
//   O      = diag(g)*Obase + P*Vhat
//   S      = (1/D_L)*S + (vscale.*Vhat)^T * Alpha   (TDM-staged operands)
// Workspace need: ~19.3 MB (S 16MB + chunk scratch).
// ---------------------------------------------------------------------------

typedef float v2f __attribute__((ext_vector_type(2)));
typedef float v8f __attribute__((ext_vector_type(8)));
typedef unsigned int u32x4 __attribute__((ext_vector_type(4)));
typedef int i32x4 __attribute__((ext_vector_type(4)));
typedef int i32x8 __attribute__((ext_vector_type(8)));

#define BDIM 4
#define TDIM 2048
#define DDIM 1024
#define LCH  64
#define NCHUNK (TDIM / LCH)
#define RENORM_T 10000.0f
#define EPSN 1e-6f

// padded LDS strides (bank-conflict rotation via TDM D# pad feature)
#define VSTR 132   // 128 payload + 4 pad dwords per row
#define ASTR 68    // 64 payload + 4 pad dwords per row

// TDM is mandatory in this version: fail loudly if the device pass lacks it.
#if defined(__HIP_DEVICE_COMPILE__) && !__has_builtin(__builtin_amdgcn_tensor_load_to_lds)
#error "gfx1250 device pass: __builtin_amdgcn_tensor_load_to_lds not available"
#endif

// workspace offsets (in floats)
static constexpr size_t OFF_S     = 0;
static constexpr size_t OFF_VHAT  = OFF_S     + (size_t)BDIM * DDIM * DDIM;
static constexpr size_t OFF_W     = OFF_VHAT  + (size_t)BDIM * LCH * DDIM;
static constexpr size_t OFF_OBASE = OFF_W     + (size_t)BDIM * LCH * DDIM;
static constexpr size_t OFF_GVV   = OFF_OBASE + (size_t)BDIM * LCH * DDIM;
static constexpr size_t OFF_GAA   = OFF_GVV   + (size_t)BDIM * LCH * LCH;
static constexpr size_t OFF_GQA   = OFF_GAA   + (size_t)BDIM * LCH * LCH;
static constexpr size_t OFF_P     = OFF_GQA   + (size_t)BDIM * LCH * LCH;
static constexpr size_t OFF_G     = OFF_P     + (size_t)BDIM * LCH * LCH;
static constexpr size_t OFF_VSC   = OFF_G     + (size_t)BDIM * LCH;
static constexpr size_t OFF_SCAL  = OFF_VSC   + (size_t)BDIM * LCH;  // 2/batch

// ---- WMMA f32 16x16x4 fragment helpers (layouts per CDNA5 ISA 7.12.2) -----
// A (16x4, MxK): lanes 0-15: M=lane, K=k0+0/1; lanes 16-31: M=lane-16, K=k0+2/3
__device__ __forceinline__ v2f ld_a(const float* __restrict__ A, int lda,
                                    int m0, int k0, int lane) {
  const int m = m0 + (lane & 15);
  const int k = k0 + ((lane >> 4) << 1);
  const float* p = A + (size_t)m * lda + k;
  v2f r; r.x = p[0]; r.y = p[1];
  return r;
}
// B (4x16, KxN) from row-major [K x N]
__device__ __forceinline__ v2f ld_b(const float* __restrict__ Bm, int ldb,
                                    int k0, int n0, int lane) {
  const int n = n0 + (lane & 15);
  const int k = k0 + ((lane >> 4) << 1);
  v2f r; r.x = Bm[(size_t)k * ldb + n]; r.y = Bm[(size_t)(k + 1) * ldb + n];
  return r;
}
// B (4x16, KxN) where B(k,n) = Y(n,k), Y row-major [N x K]  (B = Y^T)
__device__ __forceinline__ v2f ld_bT(const float* __restrict__ Y, int ldy,
                                     int k0, int n0, int lane) {
  const int n = n0 + (lane & 15);
  const int k = k0 + ((lane >> 4) << 1);
  const float* p = Y + (size_t)n * ldy + k;
  v2f r; r.x = p[0]; r.y = p[1];
  return r;
}
__device__ __forceinline__ v8f wmma_f32(v2f a, v2f b, v8f c) {
  // (neg_a, A, neg_b, B, c_mod, C, reuse_a, reuse_b)
  return __builtin_amdgcn_wmma_f32_16x16x4_f32(false, a, false, b, (short)0, c,
                                               false, false);
}
// C/D 16x16 f32: VGPR r: lanes 0-15 -> M=r, N=lane; lanes 16-31 -> M=r+8
#define TILE_MN(lane, tm, tn, mb, n)            \
  const int n  = (tn) + ((lane) & 15);          \
  const int mb = (tm) + (((lane) >> 4) << 3);

// 2D tile load global->LDS via Tensor Data Mover (D# per CDNA5 ISA).
// tile_w elems/row, tile_h rows, row stride in elements (4-byte data),
// optional LDS padding: pad_amt-coded dwords every (2<<pad_int) dwords.
__device__ __forceinline__ void tdm_load_2d(unsigned lds_off, const void* gptr,
                                            unsigned tile_w, unsigned tile_h,
                                            unsigned row_stride,
                                            unsigned pad_en, unsigned pad_int,
                                            unsigned pad_amt) {
  const unsigned long long ga = (unsigned long long)(uintptr_t)gptr;
  u32x4 g0;
  g0[0] = 1u;                                            // count=1 (valid D#)
  g0[1] = lds_off;                                       // lds_addr (bytes)
  g0[2] = (unsigned)(ga & 0xFFFFFFFFull);                // global_addr[31:0]
  g0[3] = (unsigned)((ga >> 32) & 0x01FFFFFFull) | (2u << 30);  // [56:32]|type=2
  i32x8 g1;
  g1[0] = (int)((2u << 16) | (pad_en << 20) | (pad_int << 22) | (pad_amt << 25));
  g1[1] = (int)((row_stride & 0xFFFFu) << 16);           // tensor_dim0 lo16
  g1[2] = (int)((row_stride >> 16) | ((tile_h & 0xFFFFu) << 16));  // |dim1 lo
  g1[3] = (int)((tile_h >> 16) | ((tile_w & 0xFFFFu) << 16));      // |tile_dim0
  g1[4] = (int)(tile_h & 0xFFFFu);                       // tile_dim1, dim2=0
  g1[5] = (int)row_stride;                               // dim0_stride[31:0]
  g1[6] = 0;
  g1[7] = 0;
  const i32x4 z4 = {0, 0, 0, 0};
#if defined(__clang_major__) && (__clang_major__ >= 23)
  const i32x8 z8 = {0, 0, 0, 0, 0, 0, 0, 0};
  __builtin_amdgcn_tensor_load_to_lds(g0, g1, z4, z4, z8, 0);
#else
  __builtin_amdgcn_tensor_load_to_lds(g0, g1, z4, z4, 0);
#endif
}

// ---------------------------------------------------------------------------
__global__ void k_zero(float* __restrict__ p, size_t nelem) {
  size_t i = (size_t)blockIdx.x * blockDim.x + threadIdx.x;
  if (i < nelem) p[i] = 0.f;
}

// normalize the L v-rows of this chunk: vhat = v / (||v|| + eps)
__global__ void k_prep(const float* __restrict__ v, float* __restrict__ vhat,
                       int t0) {
  __shared__ float red[256];
  const int b = blockIdx.x / LCH;
  const int t = blockIdx.x % LCH;
  const float* row = v + ((size_t)b * TDIM + (t0 + t)) * DDIM;
  float s = 0.f;
  for (int i = threadIdx.x; i < DDIM; i += 256) { float x = row[i]; s += x * x; }
  red[threadIdx.x] = s;
  __syncthreads();
  for (int off = 128; off > 0; off >>= 1) {
    if (threadIdx.x < off) red[threadIdx.x] += red[threadIdx.x + off];
    __syncthreads();
  }
  const float scale = 1.f / (sqrtf(red[0]) + EPSN);
  float* orow = vhat + ((size_t)b * LCH + t) * DDIM;
  for (int i = threadIdx.x; i < DDIM; i += 256) orow[i] = row[i] * scale;
}

// Gram matrices: Gvv = Vhat Vhat^T, Gaa = A A^T, Gqa = Q A^T (L x L each)
// 192 waves (4 batches * 3 grams * 16 tiles); 1 wave = one 16x16 tile.
__global__ void k_grams(const float* __restrict__ alpha,
                        const float* __restrict__ q,
                        const float* __restrict__ vhat,
                        float* __restrict__ gvv, float* __restrict__ gaa,
                        float* __restrict__ gqa, int t0) {
  const int lane = threadIdx.x & 31;
  const int gw = blockIdx.x * 8 + (threadIdx.x >> 5);
  const int b = gw / 48;
  const int rem = gw % 48;
  const int gram = rem / 16;
  const int tile = rem % 16;
  const int tm = (tile >> 2) << 4;
  const int tn = (tile & 3) << 4;
  const float* abase = alpha + ((size_t)b * TDIM + t0) * DDIM;
  const float* qbase = q + ((size_t)b * TDIM + t0) * DDIM;
  const float* vbase = vhat + (size_t)b * LCH * DDIM;
  const float* X; const float* Y; float* G;
  if (gram == 0)      { X = vbase; Y = vbase; G = gvv; }
  else if (gram == 1) { X = abase; Y = abase; G = gaa; }
  else                { X = qbase; Y = abase; G = gqa; }
  v8f acc = {0.f, 0.f, 0.f, 0.f, 0.f, 0.f, 0.f, 0.f};
#pragma unroll 4
  for (int k0 = 0; k0 < DDIM; k0 += 4) {
    v2f a  = ld_a(X, DDIM, tm, k0, lane);
    v2f bf = ld_bT(Y, DDIM, k0, tn, lane);
    acc = wmma_f32(a, bf, acc);
  }
  G += (size_t)b * LCH * LCH;
  TILE_MN(lane, tm, tn, mb, n);
#pragma unroll
  for (int r = 0; r < 8; ++r) G[(size_t)(mb + r) * LCH + n] = acc[r];
}

// W = Vhat * S and Obase = Q * S^T (L x d, K = d = 1024).
// Block = one (batch, which, 64-wide N strip); the S operand is staged
// k-block by k-block (64x64 f32 tiles, padded stride) with a double-buffered
// TDM pipeline: wait -> barrier -> prefetch next -> compute current.
// 8 waves: wave = one 16x32 strip (2 accumulators sharing the A fragment).
__global__ void __launch_bounds__(256)
k_ws(const float* __restrict__ q, const float* __restrict__ vhat,
     const float* __restrict__ S, float* __restrict__ W,
     float* __restrict__ Obase, int t0) {
  __shared__ float buf[2][64 * ASTR];
  const int lane = threadIdx.x & 31;
  const int wave = threadIdx.x >> 5;
  const int b = blockIdx.x / 32;
  const int rem = blockIdx.x % 32;
  const int which = rem / 16;            // 0 -> W, 1 -> Obase
  const int n0 = (rem % 16) * 64;
  const float* Sb = S + (size_t)b * DDIM * DDIM;
  const float* X = (which == 0) ? (vhat + (size_t)b * LCH * DDIM)
                                : (q + ((size_t)b * TDIM + t0) * DDIM);
  const int mt = (wave >> 1) << 4;       // M-tile base: 0,16,32,48
  const int jb = (wave & 1) << 1;        // first of this wave's two N-tiles
  v8f acc[2];
  acc[0] = (v8f){0.f, 0.f, 0.f, 0.f, 0.f, 0.f, 0.f, 0.f};
  acc[1] = (v8f){0.f, 0.f, 0.f, 0.f, 0.f, 0.f, 0.f, 0.f};

  if (threadIdx.x == 0) {
    const float* g0p = (which == 0) ? (Sb + n0) : (Sb + (size_t)n0 * DDIM);
    tdm_load_2d((unsigned)(uintptr_t)(&buf[0][0]), g0p, 64u, 64u,
                (unsigned)DDIM, 1u, 5u, 3u);
  }
  for (int ib = 0; ib < DDIM / 64; ++ib) {
    const int kb = ib * 64;
    const float* cur = &buf[ib & 1][0];
    __builtin_amdgcn_s_wait_tensorcnt(0);  // no-op for non-issuing waves
    __syncthreads();
    if (threadIdx.x == 0 && ib + 1 < DDIM / 64) {
      const int kn = kb + 64;
      const float* gnp = (which == 0) ? (Sb + (size_t)kn * DDIM + n0)
                                      : (Sb + (size_t)n0 * DDIM + kn);
      tdm_load_2d((unsigned)(uintptr_t)(&buf[(ib + 1) & 1][0]), gnp, 64u, 64u,
                  (unsigned)DDIM, 1u, 5u, 3u);
    }
    const int nn = lane & 15;
#pragma unroll 4
    for (int kk = 0; kk < 64; kk += 4) {
      v2f a = ld_a(X, DDIM, mt, kb + kk, lane);
      const int kloc = kk + ((lane >> 4) << 1);
#pragma unroll
      for (int j = 0; j < 2; ++j) {
        const int nj = 16 * (jb + j) + nn;
        v2f bf;
        if (which == 0) {           // staged tile is [k][n]
          bf.x = cur[kloc * ASTR + nj];
          bf.y = cur[(kloc + 1) * ASTR + nj];
        } else {                    // staged tile is [n][k]  (B = S^T)
          bf.x = cur[nj * ASTR + kloc];
          bf.y = cur[nj * ASTR + kloc + 1];
        }
        acc[j] = wmma_f32(a, bf, acc[j]);
      }
    }
    __syncthreads();                // reads done before buffer is overwritten
  }
  float* out = ((which == 0) ? W : Obase) + (size_t)b * LCH * DDIM;
  const int nn = lane & 15;
  const int mb = mt + ((lane >> 4) << 3);
#pragma unroll
  for (int j = 0; j < 2; ++j)
#pragma unroll
    for (int r = 0; r < 8; ++r)
      out[(size_t)(mb + r) * DDIM + n0 + 16 * (jb + j) + nn] = acc[j][r];
}

// scalar scan: divisors, D products, P / g / vscale / invDL. 1 block per batch.
__global__ void k_scan(const float* __restrict__ alpha, const float* __restrict__ W,
                       const float* __restrict__ gvv, const float* __restrict__ gaa,
                       const float* __restrict__ gqa, float* __restrict__ P,
                       float* __restrict__ g, float* __restrict__ vsc,
                       float* __restrict__ scal, int t0) {
  __shared__ float marr[LCH];
  __shared__ float Dar[LCH + 1];
  const int b = blockIdx.x;
  const int tid = threadIdx.x;
  const float* Wb = W + (size_t)b * LCH * DDIM;
  const float* ab = alpha + ((size_t)b * TDIM + t0) * DDIM;
  const float* Gv = gvv + (size_t)b * LCH * LCH;
  const float* Ga = gaa + (size_t)b * LCH * LCH;
  const float* Gq = gqa + (size_t)b * LCH * LCH;
  if (tid < LCH) {  // m_t = vhat_t^T S0 alpha_t = W_t . alpha_t
    float s = 0.f;
    const float* wr = Wb + (size_t)tid * DDIM;
    const float* ar = ab + (size_t)tid * DDIM;
    for (int j = 0; j < DDIM; ++j) s += wr[j] * ar[j];
    marr[tid] = s;
  }
  __syncthreads();
  if (tid == 0) {
    float n2 = scal[2 * b + 0];  // ||S||_F^2 carried across chunks
    Dar[0] = 1.f;
    for (int t = 0; t < LCH; ++t) {
      float u = marr[t];
      for (int s = 0; s < t; ++s)
        u += Dar[s] * Gv[t * LCH + s] * Ga[t * LCH + s];
      float cross = u / Dar[t];  // vhat_t^T S_{t-1} alpha_t
      float n2p = n2 + 2.f * cross + Gv[t * LCH + t] * Ga[t * LCH + t];
      float nrm = sqrtf(n2p);
      float dv = (nrm > RENORM_T) ? nrm : 1.f;
      n2 = n2p / (dv * dv);
      Dar[t + 1] = Dar[t] * dv;
    }
    scal[2 * b + 0] = n2;
    scal[2 * b + 1] = 1.f / Dar[LCH];
  }
  __syncthreads();
  for (int e = tid; e < LCH * LCH; e += blockDim.x) {
    int t = e / LCH, s = e % LCH;
    P[(size_t)b * LCH * LCH + e] =
        (s <= t) ? (Dar[s] / Dar[t + 1]) * Gq[e] : 0.f;
  }
  if (tid < LCH) {
    g[b * LCH + tid] = 1.f / Dar[tid + 1];
    vsc[b * LCH + tid] = Dar[tid] / Dar[LCH];
  }
}

// O = diag(g)*Obase + P*Vhat ; 1024 waves (4 batches * 256 tiles), K = L.
__global__ void k_out(const float* __restrict__ P, const float* __restrict__ vhat,
                      const float* __restrict__ Obase, const float* __restrict__ g,
                      float* __restrict__ out, int t0) {
  const int lane = threadIdx.x & 31;
  const int gw = blockIdx.x * 8 + (threadIdx.x >> 5);
  const int b = gw / 256;
  const int tile = gw % 256;
  const int tm = (tile / 64) * 16;
  const int tn = (tile % 64) * 16;
  const float* Pb = P + (size_t)b * LCH * LCH;
  const float* Vb = vhat + (size_t)b * LCH * DDIM;
  v8f acc = {0.f, 0.f, 0.f, 0.f, 0.f, 0.f, 0.f, 0.f};
#pragma unroll 4
  for (int k0 = 0; k0 < LCH; k0 += 4) {
    v2f a  = ld_a(Pb, LCH, tm, k0, lane);
    v2f bf = ld_b(Vb, DDIM, k0, tn, lane);
    acc = wmma_f32(a, bf, acc);
  }
  TILE_MN(lane, tm, tn, mb, n);
#pragma unroll
  for (int r = 0; r < 8; ++r) {
    const int m = mb + r;
    float val = acc[r] +
        g[b * LCH + m] * Obase[((size_t)b * LCH + m) * DDIM + n];
    out[((size_t)b * TDIM + t0 + m) * DDIM + n] = val;
  }
}

// S = invDL*S + (vscale .* Vhat)^T * Alpha  (d x d accumulate, K = L).
// Block = 128x64 region of S; operands staged to LDS by TDM (padded strides),
// 8 waves each compute a 16x64 strip with 4 accumulators fed from ds_loads.
__global__ void __launch_bounds__(256)
k_supd(const float* __restrict__ alpha, const float* __restrict__ vhat,
       const float* __restrict__ vsc, const float* __restrict__ scal,
       float* __restrict__ S, int t0) {
  __shared__ float ldsV[LCH * VSTR];  // vhat[s][i0..i0+127], padded rows
  __shared__ float ldsA[LCH * ASTR];  // alpha[s][n0..n0+63], padded rows
  const int lane = threadIdx.x & 31;
  const int wave = threadIdx.x >> 5;
  const int blocksPerB = (DDIM / 128) * (DDIM / 64);   // 128
  const int b = blockIdx.x / blocksPerB;
  const int rem = blockIdx.x % blocksPerB;
  const int i0 = (rem / (DDIM / 64)) * 128;            // M block start
  const int n0 = (rem % (DDIM / 64)) * 64;             // N block start
  const float* Vb = vhat + (size_t)b * LCH * DDIM;
  const float* ab = alpha + ((size_t)b * TDIM + t0) * DDIM;
  const float* vs = vsc + b * LCH;

  if (threadIdx.x == 0) {
    // pad codes: interval 6 -> every 128 dwords, 5 -> every 64; amount 3 -> 4
    tdm_load_2d((unsigned)(uintptr_t)(&ldsV[0]), Vb + i0, 128u, (unsigned)LCH,
                (unsigned)DDIM, 1u, 6u, 3u);
    tdm_load_2d((unsigned)(uintptr_t)(&ldsA[0]), ab + n0, 64u, (unsigned)LCH,
                (unsigned)DDIM, 1u, 5u, 3u);
  }
  __builtin_amdgcn_s_wait_tensorcnt(0);   // no-op for non-issuing waves
  __syncthreads();
  // fold the per-step scale into the staged vhat rows
  for (int idx = threadIdx.x; idx < LCH * 128; idx += 256)
    ldsV[(idx >> 7) * VSTR + (idx & 127)] *= vs[idx >> 7];
  __syncthreads();

  const float invDL = scal[2 * b + 1];
  float* Sb = S + (size_t)b * DDIM * DDIM;
  const int mi = (wave << 4) + (lane & 15);            // row in ldsV payload
  const int nn = lane & 15;
  const int mb = (wave << 4) + ((lane >> 4) << 3);     // local C row base
  v8f acc[4];
#pragma unroll
  for (int j = 0; j < 4; ++j)
#pragma unroll
    for (int r = 0; r < 8; ++r)
      acc[j][r] = invDL * Sb[(size_t)(i0 + mb + r) * DDIM + n0 + 16 * j + nn];
#pragma unroll 2
  for (int s0 = 0; s0 < LCH; s0 += 4) {
    const int s = s0 + ((lane >> 4) << 1);
    v2f a;   // A(m=i, k=s) = vscale[s] * vhat[s][i]  (pre-scaled in LDS)
    a.x = ldsV[s * VSTR + mi];
    a.y = ldsV[(s + 1) * VSTR + mi];
#pragma unroll
    for (int j = 0; j < 4; ++j) {
      v2f bf;
      bf.x = ldsA[s * ASTR + 16 * j + nn];
      bf.y = ldsA[(s + 1) * ASTR + 16 * j + nn];
      acc[j] = wmma_f32(a, bf, acc[j]);
    }
  }
#pragma unroll
  for (int j = 0; j < 4; ++j)
#pragma unroll
    for (int r = 0; r < 8; ++r)
      Sb[(size_t)(i0 + mb + r) * DDIM + n0 + 16 * j + nn] = acc[j][r];
}

// ---------------------------------------------------------------------------
extern "C" void kernel_launch(void* const* d_in, const int* in_sizes, int n_in,
                              void* d_out, int out_size, void* d_ws,
                              size_t ws_size, hipStream_t stream) {
  const float* v     = (const float*)d_in[0];
  const float* alpha = (const float*)d_in[1];
  const float* q     = (const float*)d_in[2];
  float* out = (float*)d_out;
  float* ws  = (float*)d_ws;

  float* S     = ws + OFF_S;
  float* Vhat  = ws + OFF_VHAT;
  float* W     = ws + OFF_W;
  float* Obase = ws + OFF_OBASE;
  float* Gvv   = ws + OFF_GVV;
  float* Gaa   = ws + OFF_GAA;
  float* Gqa   = ws + OFF_GQA;
  float* P     = ws + OFF_P;
  float* g     = ws + OFF_G;
  float* vsc   = ws + OFF_VSC;
  float* scal  = ws + OFF_SCAL;

  {  // S = 0, carried ||S||^2 = 0
    const size_t n = (size_t)BDIM * DDIM * DDIM;
    k_zero<<<(unsigned)((n + 255) / 256), 256, 0, stream>>>(S, n);
    k_zero<<<1, 32, 0, stream>>>(scal, (size_t)2 * BDIM);
  }

  for (int c = 0; c < NCHUNK; ++c) {
    const int t0 = c * LCH;
    k_prep <<<BDIM * LCH, 256, 0, stream>>>(v, Vhat, t0);
    k_grams<<<24,         256, 0, stream>>>(alpha, q, Vhat, Gvv, Gaa, Gqa, t0);
    k_ws   <<<BDIM * 32,  256, 0, stream>>>(q, Vhat, S, W, Obase, t0);
    k_scan <<<BDIM,       256, 0, stream>>>(alpha, W, Gvv, Gaa, Gqa, P, g, vsc,
                                            scal, t0);
    k_out  <<<128,        256, 0, stream>>>(P, Vhat, Obase, g, out, t0);
    k_supd <<<BDIM * 128, 256, 0, stream>>>(alpha, Vhat, vsc, scal, S, t0);
  }
}